// Conjunction_57793079935283
// MI455X (gfx1250) — compile-verified
//
#include <hip/hip_runtime.h>

typedef __attribute__((ext_vector_type(2))) float v2f;
typedef __attribute__((ext_vector_type(8))) float v8f;

#define BATCH 128
#define IN_F  512
#define OUT_F 1024
#define RROWS 16    // distinct weight rows actually used (ROW_IDX = j % 16)
#define NGRP  32
#define DELTA 0.1f

// ---------------------------------------------------------------------------
// Phase 1: collapse x (128x512) into three 128x16 matrices:
//   P[b,r] = sum_g x[b,16g+r] * (x >= -1)      (masked sum)
//   S[b,r] = sum_g |x[b,16g+r]|                (abs sum)
//   M[b,r] = max_g |x[b,16g+r]|                (abs max)
// One wave per batch row; lane l covers r = l%16, half h = l/16 strides groups.
// ---------------------------------------------------------------------------
__global__ __launch_bounds__(32)
void phase1_reduce(const float* __restrict__ x,
                   float* __restrict__ P,
                   float* __restrict__ S,
                   float* __restrict__ M) {
    const int b = blockIdx.x;
    const int l = threadIdx.x;
    const int r = l & 15;
    const int h = l >> 4;
    const float* xb = x + b * IN_F;

    float p = 0.0f, s = 0.0f, m = 0.0f;
    for (int g = h; g < NGRP; g += 2) {
        float v = xb[g * RROWS + r];
        float a = fabsf(v);
        p += (v >= -1.0f) ? v : 0.0f;
        s += a;
        m = fmaxf(m, a);
    }
    // combine the two halves of the wave (wave32)
    p += __shfl_xor(p, 16, 32);
    s += __shfl_xor(s, 16, 32);
    m = fmaxf(m, __shfl_xor(m, 16, 32));

    if (l < 16) {
        P[b * RROWS + r] = p;
        S[b * RROWS + r] = s;
        M[b * RROWS + r] = m;
    }
}

// ---------------------------------------------------------------------------
// Phase 2: out[b,o] = (P @ W)[b,o] + DELTA * ( max_r(|W[r,o]|*M[b,r])
//                                              - (S @ |W|)[b,o] )
// W = weights rows 0..15. Two K=16 fp32 GEMMs via V_WMMA_F32_16X16X4_F32
// (4 WMMAs each), plus a VALU max loop in C-matrix register layout.
// Block: 8 waves; block covers one 16-row batch tile x 128 out columns,
// wave w owns one 16x16 output tile.
// ---------------------------------------------------------------------------
__global__ __launch_bounds__(256)
void phase2_wmma(const float* __restrict__ weights,
                 const float* __restrict__ P,
                 const float* __restrict__ S,
                 const float* __restrict__ M,
                 float* __restrict__ out) {
    __shared__ float lP[16 * 16];
    __shared__ float lS[16 * 16];
    __shared__ float lM[16 * 16];
    __shared__ float lW[8][16 * 16];   // per-wave W tile   [r][n]
    __shared__ float lA[8][16 * 16];   // per-wave |W| tile [r][n]

    const int tid   = threadIdx.x;
    const int w     = tid >> 5;
    const int l     = tid & 31;
    const int btile = blockIdx.x;                  // 0..7
    const int obase = (blockIdx.y * 8 + w) * 16;   // this wave's out column base

    // stage P/S/M tiles (shared by all 8 waves): one element per thread
    {
        int row = tid >> 4, col = tid & 15;
        int idx = (btile * 16 + row) * RROWS + col;
        lP[tid] = P[idx];
        lS[tid] = S[idx];
        lM[tid] = M[idx];
    }
    // stage this wave's 16x16 weight tile (+abs)
    for (int i = l; i < 256; i += 32) {
        int r = i >> 4, n = i & 15;
        float wv = weights[r * OUT_F + obase + n];
        lW[w][i] = wv;
        lA[w][i] = fabsf(wv);
    }
    __syncthreads();

    const int h = l >> 4;     // half of wave
    const int m = l & 15;     // A-matrix row (batch within tile)
    const int n = l & 15;     // B/C-matrix column

    // --- two K=16 GEMMs, 4 x v_wmma_f32_16x16x4_f32 each ---
    // A layout (ISA 7.12.2): VGPR0 = K0 (lanes 0-15) / K2 (lanes 16-31),
    //                        VGPR1 = K1 / K3.  B follows the same pattern.
    v8f c1 = {};  // P @ W
    v8f c2 = {};  // S @ |W|
#pragma unroll
    for (int k0 = 0; k0 < RROWS; k0 += 4) {
        const int ka = k0 + 2 * h;
        v2f a1, a2, b1, b2;
        a1.x = lP[m * 16 + ka];         a1.y = lP[m * 16 + ka + 1];
        a2.x = lS[m * 16 + ka];         a2.y = lS[m * 16 + ka + 1];
        b1.x = lW[w][ka * 16 + n];      b1.y = lW[w][(ka + 1) * 16 + n];
        b2.x = lA[w][ka * 16 + n];      b2.y = lA[w][(ka + 1) * 16 + n];
        c1 = __builtin_amdgcn_wmma_f32_16x16x4_f32(false, a1, false, b1,
                                                   (short)0, c1, false, false);
        c2 = __builtin_amdgcn_wmma_f32_16x16x4_f32(false, a2, false, b2,
                                                   (short)0, c2, false, false);
    }

    // --- max_r |W[r,o]| * M[b,r], computed directly in the C VGPR layout:
    // VGPR v holds (row = mbase+v, col = n). LDS reads are half-wave
    // broadcasts (conflict-free). Products are >= 0 so init 0 is exact.
    const int mbase = h * 8;
    float mx[8];
#pragma unroll
    for (int v = 0; v < 8; ++v) mx[v] = 0.0f;
    for (int r = 0; r < RROWS; ++r) {
        float wa = lA[w][r * 16 + n];
#pragma unroll
        for (int v = 0; v < 8; ++v)
            mx[v] = fmaxf(mx[v], wa * lM[(mbase + v) * 16 + r]);
    }

    // --- combine and store: out = C1 + DELTA * (max - C2)
#pragma unroll
    for (int v = 0; v < 8; ++v) {
        int row = btile * 16 + mbase + v;
        out[row * OUT_F + obase + n] = c1[v] + DELTA * (mx[v] - c2[v]);
    }
}

extern "C" void kernel_launch(void* const* d_in, const int* in_sizes, int n_in,
                              void* d_out, int out_size, void* d_ws, size_t ws_size,
                              hipStream_t stream) {
    (void)in_sizes; (void)n_in; (void)out_size; (void)ws_size;
    const float* x       = (const float*)d_in[0];   // (128, 512)
    const float* weights = (const float*)d_in[1];   // (32, 1024), rows 0..15 used
    float* out = (float*)d_out;                     // (128, 1024)

    float* P = (float*)d_ws;            // 128*16
    float* S = P + BATCH * RROWS;       // 128*16
    float* M = S + BATCH * RROWS;       // 128*16  (24 KB total)

    phase1_reduce<<<dim3(BATCH), dim3(32), 0, stream>>>(x, P, S, M);
    phase2_wmma<<<dim3(8, 8), dim3(256), 0, stream>>>(weights, P, S, M, out);
}